// PaiPool_35064113005154
// MI455X (gfx1250) — compile-verified
//
#include <hip/hip_runtime.h>
#include <hip/hip_bf16.h>

typedef float v2f __attribute__((ext_vector_type(2)));
typedef float v8f __attribute__((ext_vector_type(8)));

#define NPTS   8192
#define NPOOL  2048
#define KSAMP  32
#define CFEAT  64
#define RAD2   0.16f   // 0.4^2

// -------------------------------------------------------------------------
// Kernel 1: farthest point sampling. One block per batch, 1024 threads,
// each thread owns 8 contiguous points held entirely in registers.
// argmax with first-index tie-break packed as u64 (distbits<<32)|(~idx).
// -------------------------------------------------------------------------
__global__ void __launch_bounds__(1024)
fps_kernel(const float* __restrict__ x, int* __restrict__ fps_idx,
           int N, int M)
{
    __shared__ float cent[3];
    __shared__ unsigned long long wkeys[32];
    __shared__ int s_far;

    const int t = threadIdx.x;
    const int b = blockIdx.x;
    const float* xb = x + (size_t)b * 3 * N;

    float px[8], py[8], pz[8], dist[8];
    const int base = t * 8;
#pragma unroll
    for (int j = 0; j < 8; ++j) {
        px[j] = xb[base + j];
        py[j] = xb[N + base + j];
        pz[j] = xb[2 * N + base + j];
        dist[j] = __builtin_inff();
    }

    int far = 0;
    for (int i = 0; i < M; ++i) {
        if (t == 0) fps_idx[b * M + i] = far;
        if ((far >> 3) == t) {
            int j = far & 7;
            cent[0] = px[j]; cent[1] = py[j]; cent[2] = pz[j];
        }
        __syncthreads();
        const float cx = cent[0], cy = cent[1], cz = cent[2];

        unsigned long long key = 0ull;
#pragma unroll
        for (int j = 0; j < 8; ++j) {
            float dx = px[j] - cx, dy = py[j] - cy, dz = pz[j] - cz;
            float d = dx * dx + dy * dy + dz * dz;
            dist[j] = fminf(dist[j], d);
            unsigned long long k2 =
                ((unsigned long long)__float_as_uint(dist[j]) << 32) |
                (unsigned)(0xFFFFFFFFu - (unsigned)(base + j));
            key = (k2 > key) ? k2 : key;
        }
        // wave-level max
#pragma unroll
        for (int off = 1; off < 32; off <<= 1) {
            unsigned long long o = __shfl_xor(key, off, 32);
            key = (o > key) ? o : key;
        }
        const int lane = t & 31, w = t >> 5;
        if (lane == 0) wkeys[w] = key;
        __syncthreads();
        if (t < 32) {
            key = wkeys[t];
#pragma unroll
            for (int off = 1; off < 32; off <<= 1) {
                unsigned long long o = __shfl_xor(key, off, 32);
                key = (o > key) ? o : key;
            }
            if (t == 0) s_far = (int)(0xFFFFFFFFu - (unsigned)key);
        }
        __syncthreads();
        far = s_far;
    }
}

// -------------------------------------------------------------------------
// Kernel 2: ball query (first 32 in-radius indices, ascending, padded with
// first found) fused with x_out = max over group of coordinates.
// One wave (32 lanes) per center; 8 waves per block.
// -------------------------------------------------------------------------
__global__ void __launch_bounds__(256)
ballq_kernel(const float* __restrict__ x, const int* __restrict__ fps_idx,
             int* __restrict__ sub_idx, float* __restrict__ x_out,
             int N, int M)
{
    const int w = threadIdx.x >> 5, lane = threadIdx.x & 31;
    const int g = blockIdx.x * 8 + w;           // center id in [0, B*M)
    const int b = g / M, m = g % M;
    const float* xb = x + (size_t)b * 3 * N;

    const int ci = fps_idx[g];
    const float cx = xb[ci], cy = xb[N + ci], cz = xb[2 * N + ci];

    int cnt = 0, first = -1, sel = -1;
    for (int bas = 0; bas < N && cnt < KSAMP; bas += 32) {
        const int p = bas + lane;
        float dx = xb[p] - cx, dy = xb[N + p] - cy, dz = xb[2 * N + p] - cz;
        unsigned mask = (unsigned)__ballot(dx * dx + dy * dy + dz * dz < RAD2);
        if (mask) {
            if (first < 0) first = bas + __builtin_ctz(mask);
            const int pc = __popc(mask);
            const int slot = lane;
            if (slot >= cnt && slot < cnt + pc && slot < KSAMP) {
                int j = slot - cnt;
                unsigned mm = mask;
                for (int q = 0; q < j; ++q) mm &= mm - 1;
                sel = bas + __builtin_ctz(mm);
            }
            cnt += pc;
        }
    }
    if (cnt > KSAMP) cnt = KSAMP;
    if (first < 0) first = N - 1;               // never hit: center is a data point
    if (lane >= cnt || sel < 0) sel = first;

    sub_idx[(size_t)g * KSAMP + lane] = sel;

    // x_out: max over the 32 selected points of each coordinate
    float vx = xb[sel], vy = xb[N + sel], vz = xb[2 * N + sel];
#pragma unroll
    for (int off = 1; off < 32; off <<= 1) {
        vx = fmaxf(vx, __shfl_xor(vx, off, 32));
        vy = fmaxf(vy, __shfl_xor(vy, off, 32));
        vz = fmaxf(vz, __shfl_xor(vz, off, 32));
    }
    if (lane == 0) {
        x_out[((size_t)b * 3 + 0) * M + m] = vx;
        x_out[((size_t)b * 3 + 1) * M + m] = vy;
        x_out[((size_t)b * 3 + 2) * M + m] = vz;
    }
}

// -------------------------------------------------------------------------
// Kernel 3: per-center O = W(64x64) @ F(64x32), then rowmax over 32 samples.
// fp32 WMMA (V_WMMA_F32_16X16X4_F32). One block (4 waves) per center:
// gather F into LDS, each wave owns a 16-row tile of O and both 16-col
// N-tiles (two accumulators, 32 WMMA ops/wave).
// -------------------------------------------------------------------------
__global__ void __launch_bounds__(128)
gemm_max_kernel(const float* __restrict__ feature,
                const float* __restrict__ weight,
                const int* __restrict__ sub_idx,
                float* __restrict__ f_out,
                int N, int M)
{
    __shared__ float Fl[CFEAT][KSAMP];   // gathered feature tile (64x32)
    __shared__ int Sidx[KSAMP];

    const int t = threadIdx.x;
    const int g = blockIdx.x;            // center id
    const int b = g / M, m = g % M;

    if (t < KSAMP) {
        int v = sub_idx[(size_t)g * KSAMP + t];
        Sidx[t] = (v < N) ? v : (N - 1);
    }
    __syncthreads();

    const float* fb = feature + (size_t)b * CFEAT * N;
    for (int e = t; e < CFEAT * KSAMP; e += 128) {
        const int c = e >> 5, s = e & 31;
        Fl[c][s] = fb[(size_t)c * N + Sidx[s]];
    }
    __syncthreads();

    const int w = t >> 5, lane = t & 31;
    const int Mo = w * 16;               // output row tile
    const int h = lane >> 4;             // half selects K pair
    const int n = lane & 15;             // N (sample) / M-row index within tile

    v8f c0 = {0.f, 0.f, 0.f, 0.f, 0.f, 0.f, 0.f, 0.f};
    v8f c1 = {0.f, 0.f, 0.f, 0.f, 0.f, 0.f, 0.f, 0.f};

    const int row = Mo + n;              // A-matrix row (lane = M % 16)
#pragma unroll
    for (int k0 = 0; k0 < CFEAT; k0 += 4) {
        const int ka = k0 + 2 * h;       // this half's K pair
        v2f a, b0, b1;
        a.x = weight[row * CFEAT + ka];
        a.y = weight[row * CFEAT + ka + 1];
        b0.x = Fl[ka][n];      b0.y = Fl[ka + 1][n];
        b1.x = Fl[ka][16 + n]; b1.y = Fl[ka + 1][16 + n];
        c0 = __builtin_amdgcn_wmma_f32_16x16x4_f32(
                false, a, false, b0, (short)0, c0, false, false);
        c1 = __builtin_amdgcn_wmma_f32_16x16x4_f32(
                false, a, false, b1, (short)0, c1, false, false);
    }

    // max over both N-tiles, then butterfly-max over the 16 lanes of a half
    float d[8];
#pragma unroll
    for (int i = 0; i < 8; ++i) d[i] = fmaxf(c0[i], c1[i]);
#pragma unroll
    for (int off = 1; off < 16; off <<= 1) {
#pragma unroll
        for (int i = 0; i < 8; ++i)
            d[i] = fmaxf(d[i], __shfl_xor(d[i], off, 32));
    }
    // lanes 0 and 16 hold rows Mo..Mo+7 and Mo+8..Mo+15 respectively
    if (n == 0) {
        const int rbase = Mo + h * 8;
#pragma unroll
        for (int i = 0; i < 8; ++i)
            f_out[((size_t)b * CFEAT + rbase + i) * M + m] = d[i];
    }
}

extern "C" void kernel_launch(void* const* d_in, const int* in_sizes, int n_in,
                              void* d_out, int out_size, void* d_ws, size_t ws_size,
                              hipStream_t stream) {
    (void)in_sizes; (void)n_in; (void)out_size; (void)ws_size;
    const float* x       = (const float*)d_in[0];   // (8,3,8192)
    const float* feature = (const float*)d_in[1];   // (8,64,8192)
    const float* weight  = (const float*)d_in[2];   // (64,64)
    // d_in[3] = num_pool (2048), fixed by problem shapes

    const int B = 8, N = NPTS, M = NPOOL;

    int* fps_idx = (int*)d_ws;                      // B*M ints
    int* sub_idx = fps_idx + B * M;                 // B*M*32 ints
    float* x_out = (float*)d_out;                   // B*3*M
    float* f_out = x_out + (size_t)B * 3 * M;       // B*64*M

    fps_kernel<<<B, 1024, 0, stream>>>(x, fps_idx, N, M);
    ballq_kernel<<<(B * M) / 8, 256, 0, stream>>>(x, fps_idx, sub_idx, x_out, N, M);
    gemm_max_kernel<<<B * M, 128, 0, stream>>>(feature, weight, sub_idx, f_out, N, M);
}